// MessagePassing_34857954574420
// MI455X (gfx1250) — compile-verified
//
#include <hip/hip_runtime.h>

// MessagePassing scatter-add: out[dst[e]] += x[src[e]], E=800k, D=64, N=50k, f32.
// Zero matmul FLOPs -> WMMA not applicable; op is HBM/L2-atomic bound
// (~430MB traffic ~= 20us @ 23.3 TB/s). CDNA5 paths: double-buffered
// tensor_load_to_lds (TDM) + s_wait_tensorcnt pipeline + global prefetch +
// native global_atomic_add_f32.

#define CHUNK   512             // edges per TDM chunk
#define THREADS 256             // 8 wave32 waves
#define WAVES   (THREADS / 32)
#define D_FEAT  64

typedef __attribute__((ext_vector_type(4))) unsigned int v4u;
typedef __attribute__((ext_vector_type(8))) int          v8i;
typedef __attribute__((ext_vector_type(4))) int          v4i;

#if defined(__AMDGCN__) && __has_builtin(__builtin_amdgcn_tensor_load_to_lds) && \
    __has_builtin(__builtin_amdgcn_s_wait_tensorcnt)
#define USE_TDM 1
#else
#define USE_TDM 0
#endif

#if USE_TDM
// Issue one TDM descriptor: 2-row tile (n x 2 int32, row stride E elements)
// from gsrc (src row; dst row at gsrc + E) into LDS at lds_off.
// LDS receives n src indices followed immediately by n dst indices.
__device__ __forceinline__ void tdm_load_idx(const int* gsrc, unsigned lds_off,
                                             int n, int E) {
  unsigned long long ga = (unsigned long long)(size_t)gsrc;
  unsigned un = (unsigned)n;
  // D# group 0: count=1 | lds_addr | global_addr[56:0] | type=2
  v4u g0 = { 1u, lds_off, (unsigned)ga,
             (unsigned)((ga >> 32) & 0x01FFFFFFull) | 0x80000000u };
  // D# group 1: data_size=2 (4B); tensor_dim0=n; tensor_dim1=2; tile_dim0=n;
  //             tile_dim1=2; tile_dim2=0; tensor_dim0_stride=E
  v8i g1 = { (int)(2u << 16),                              // wg_mask=0 | data_size=2
             (int)((un & 0xFFFFu) << 16),                  // bar_addr=0 | tdim0.lo16
             (int)(((un >> 16) & 0xFFFFu) | (2u << 16)),   // tdim0.hi16 | tdim1.lo16=2
             (int)((un & 0xFFFFu) << 16),                  // tdim1.hi16=0 | tile_dim0
             2,                                            // tile_dim1=2 | tile_dim2=0
             E,                                            // tdim0_stride.lo32
             0, 0 };
  v4i gz = {0, 0, 0, 0};
#if __clang_major__ >= 23
  v8i gz8 = {0, 0, 0, 0, 0, 0, 0, 0};
  __builtin_amdgcn_tensor_load_to_lds(g0, g1, gz, gz, gz8, 0);
#else
  __builtin_amdgcn_tensor_load_to_lds(g0, g1, gz, gz, 0);
#endif
}
#endif

// One wave per edge: 32 lanes x float2 = 64 features.
// Coalesced 256B row load, 2 native f32 atomic adds per lane.
__device__ __forceinline__ void process_chunk(const int* s_buf, int n,
                                              const float* __restrict__ x,
                                              float* __restrict__ out,
                                              int wave, int lane) {
  for (int e = wave; e < n; e += WAVES) {
    const int src = s_buf[e];
    const int dst = s_buf[n + e];
    const int e2  = e + WAVES;
    if (e2 < n)  // prefetch next edge's source row (global_prefetch_b8 path)
      __builtin_prefetch(x + (size_t)s_buf[e2] * D_FEAT + lane * 2, 0, 0);
    const float2 v = ((const float2*)(x + (size_t)src * D_FEAT))[lane];
    float* o = out + (size_t)dst * D_FEAT + lane * 2;
    unsafeAtomicAdd(o,     v.x);
    unsafeAtomicAdd(o + 1, v.y);
  }
}

__global__ __launch_bounds__(THREADS) void scatter_add_kernel(
    const float* __restrict__ x,
    const int*   __restrict__ edge,   // [2, E] flat: src row then dst row
    float*       __restrict__ out,
    int E) {
  __shared__ int s_idx[2][2 * CHUNK];           // double buffer, 8 KB total
  const int base = blockIdx.x * (2 * CHUNK);
  const int rem  = E - base;
  if (rem <= 0) return;                          // block-uniform
  const int n0 = rem < CHUNK ? rem : CHUNK;
  const int r1 = rem - CHUNK;
  const int n1 = r1 < CHUNK ? r1 : CHUNK;        // may be <= 0
  const int wave = threadIdx.x >> 5;
  const int lane = threadIdx.x & 31;

#if USE_TDM
  if (threadIdx.x < 32) {                        // wave 0 drives the DMA pipeline
    tdm_load_idx(edge + base, (unsigned)(size_t)&s_idx[0][0], n0, E);
    if (n1 > 0) {
      tdm_load_idx(edge + base + CHUNK, (unsigned)(size_t)&s_idx[1][0], n1, E);
      __builtin_amdgcn_s_wait_tensorcnt(1);      // chunk0 done; chunk1 in flight
    } else {
      __builtin_amdgcn_s_wait_tensorcnt(0);
    }
  }
  __syncthreads();
#else
  for (int i = threadIdx.x; i < n0; i += THREADS) {
    s_idx[0][i]      = edge[base + i];
    s_idx[0][n0 + i] = edge[E + base + i];
  }
  for (int i = threadIdx.x; i < n1; i += THREADS) {
    s_idx[1][i]      = edge[base + CHUNK + i];
    s_idx[1][n1 + i] = edge[E + base + CHUNK + i];
  }
  __syncthreads();
#endif

  process_chunk(s_idx[0], n0, x, out, wave, lane);

  if (n1 > 0) {                                  // block-uniform
#if USE_TDM
    if (threadIdx.x < 32) __builtin_amdgcn_s_wait_tensorcnt(0);
    __syncthreads();
#endif
    process_chunk(s_idx[1], n1, x, out, wave, lane);
  }
}

__global__ void zero_out_kernel(float4* __restrict__ out, int n4) {
  int i = blockIdx.x * blockDim.x + threadIdx.x;
  if (i < n4) out[i] = make_float4(0.f, 0.f, 0.f, 0.f);
}

extern "C" void kernel_launch(void* const* d_in, const int* in_sizes, int n_in,
                              void* d_out, int out_size, void* d_ws, size_t ws_size,
                              hipStream_t stream) {
  const float* x    = (const float*)d_in[0];
  const int*   edge = (const int*)d_in[1];   // int inputs arrive as int32 per harness
  float*       out  = (float*)d_out;
  const int E  = in_sizes[1] / 2;            // 800000
  const int n4 = out_size / 4;               // 50000*64/4 float4 stores

  zero_out_kernel<<<(n4 + 255) / 256, 256, 0, stream>>>((float4*)out, n4);

  const int grid = (E + 2 * CHUNK - 1) / (2 * CHUNK);  // 782 blocks
  scatter_add_kernel<<<grid, THREADS, 0, stream>>>(x, edge, out, E);
}